// FastAttention_48696339202315
// MI455X (gfx1250) — compile-verified
//
#include <hip/hip_runtime.h>

typedef __attribute__((ext_vector_type(2))) float v2f;
typedef __attribute__((ext_vector_type(8))) float v8f;

#define BH   32      // B*H
#define NSEQ 4096
#define DIM  64      // D == F
#define CHUNK 512    // rows of K handled per pass-1 workgroup

// ---------------------------------------------------------------------------
// zero the reduction workspace (k_sum[32][64] and ctx[32][64])
// ---------------------------------------------------------------------------
__global__ void FA_zero_ws(float* __restrict__ ws, int n) {
    int i = blockIdx.x * blockDim.x + threadIdx.x;
    if (i < n) ws[i] = 0.0f;
}

// ---------------------------------------------------------------------------
// Pass 1: k_prime = exp(K @ P^T); k_sum[f] = sum_n k_prime;
//         ctx[f]  = sum_n k_prime[n,f] * v[n,f]
// One workgroup = one (bh, 512-row chunk). 8 waves: wave = (rowGroup<<2)|colTile
// ---------------------------------------------------------------------------
__global__ __launch_bounds__(256) void FA_pass1(
    const float* __restrict__ K, const float* __restrict__ V,
    const float* __restrict__ P,
    float* __restrict__ ksum_ws, float* __restrict__ ctx_ws)
{
    __shared__ float sP[DIM * DIM];       // proj staged once: 16 KB
    __shared__ float sred[2][8][32];      // cross-wave reduction

    const int bh    = blockIdx.x;         // 0..31
    const int chunk = blockIdx.y;         // 0..7
    const int tid   = threadIdx.x;

    for (int i = tid; i < DIM * DIM; i += 256) sP[i] = P[i];
    __syncthreads();

    const int wave = tid >> 5;
    const int lane = tid & 31;
    const int l15  = lane & 15;
    const int hi   = lane >> 4;           // half-wave select
    const int ct   = wave & 3;            // column tile 0..3 (f = 16*ct + l15)
    const int rg   = wave >> 2;           // row group 0/1
    const int col  = ct * 16 + l15;       // this lane's f column in C layout

    const float* Kbh = K + (size_t)bh * NSEQ * DIM;
    const float* Vbh = V + (size_t)bh * NSEQ * DIM;

    float ksum_p = 0.0f, ctx_p = 0.0f;

    const int nBase = chunk * CHUNK;
    for (int n0 = nBase + rg * 16; n0 < nBase + CHUNK; n0 += 32) {
        v8f c = {};
        // A fragment rows: M = l15; VGPR0/1 hold K = k0+2*hi, k0+2*hi+1
        const float* Arow = Kbh + (size_t)(n0 + l15) * DIM;
        #pragma unroll
        for (int k0 = 0; k0 < DIM; k0 += 4) {
            v2f a, b;
            a.x = Arow[k0 + 2 * hi];
            a.y = Arow[k0 + 2 * hi + 1];
            // B = P^T : B[k][f] = P[f*64 + k], same half-wave K split
            b.x = sP[col * DIM + k0 + 2 * hi];
            b.y = sP[col * DIM + k0 + 2 * hi + 1];
            c = __builtin_amdgcn_wmma_f32_16x16x4_f32(
                    false, a, false, b, (short)0, c, false, false);
        }
        // C layout: lane holds column `col`, rows m = n0 + r + 8*hi
        const float* Vcol = Vbh + (size_t)(n0 + 8 * hi) * DIM + col;
        #pragma unroll
        for (int r = 0; r < 8; ++r) {
            float e = __expf(c[r]);
            ksum_p += e;
            ctx_p  += e * Vcol[(size_t)r * DIM];
        }
    }

    sred[0][wave][lane] = ksum_p;
    sred[1][wave][lane] = ctx_p;
    __syncthreads();

    if (tid < 64) {                       // one thread per f
        const int f   = tid;
        const int ctf = f >> 4;
        const int lf  = f & 15;
        float ks = sred[0][ctf][lf] + sred[0][ctf][lf + 16]
                 + sred[0][ctf + 4][lf] + sred[0][ctf + 4][lf + 16];
        float cx = sred[1][ctf][lf] + sred[1][ctf][lf + 16]
                 + sred[1][ctf + 4][lf] + sred[1][ctf + 4][lf + 16];
        atomicAdd(&ksum_ws[bh * DIM + f], ks);
        atomicAdd(&ctx_ws[bh * DIM + f], cx);
    }
}

// ---------------------------------------------------------------------------
// Pass 2: q_prime = exp(Q @ P^T); d_inv = 1/(q_prime . (k_sum+eps));
//         out = q_prime * ctx * d_inv
// One wave = 16 rows x 64 cols (4 WMMA chains). WG = 8 waves = 128 rows.
// ---------------------------------------------------------------------------
__global__ __launch_bounds__(256) void FA_pass2(
    const float* __restrict__ Q, const float* __restrict__ P,
    const float* __restrict__ ksum_ws, const float* __restrict__ ctx_ws,
    float* __restrict__ out)
{
    __shared__ float sP[DIM * DIM];

    const int bh   = blockIdx.x;          // 0..31
    const int slab = blockIdx.y;          // 0..31 (128 rows each)
    const int tid  = threadIdx.x;

    for (int i = tid; i < DIM * DIM; i += 256) sP[i] = P[i];
    __syncthreads();

    const int wave = tid >> 5;
    const int lane = tid & 31;
    const int l15  = lane & 15;
    const int hi   = lane >> 4;
    const int n0   = slab * 128 + wave * 16;

    const float* Qbh = Q + (size_t)bh * NSEQ * DIM;

    float ks[4], cx[4];
    #pragma unroll
    for (int ct = 0; ct < 4; ++ct) {
        ks[ct] = ksum_ws[bh * DIM + ct * 16 + l15] + 1e-6f;
        cx[ct] = ctx_ws [bh * DIM + ct * 16 + l15];
    }

    v8f c[4] = {};
    const float* Arow = Qbh + (size_t)(n0 + l15) * DIM;
    #pragma unroll
    for (int k0 = 0; k0 < DIM; k0 += 4) {
        v2f a;
        a.x = Arow[k0 + 2 * hi];
        a.y = Arow[k0 + 2 * hi + 1];
        #pragma unroll
        for (int ct = 0; ct < 4; ++ct) {
            v2f b;
            b.x = sP[(ct * 16 + l15) * DIM + k0 + 2 * hi];
            b.y = sP[(ct * 16 + l15) * DIM + k0 + 2 * hi + 1];
            c[ct] = __builtin_amdgcn_wmma_f32_16x16x4_f32(
                        false, a, false, b, (short)0, c[ct], false, false);
        }
    }

    // exp + per-row denominator (rows m = n0 + r + 8*hi)
    float e[4][8];
    float rinv[8];
    #pragma unroll
    for (int r = 0; r < 8; ++r) {
        float p = 0.0f;
        #pragma unroll
        for (int ct = 0; ct < 4; ++ct) {
            e[ct][r] = __expf(c[ct][r]);
            p += e[ct][r] * ks[ct];
        }
        // butterfly over the 16 lanes of this half-wave (masks < 16 stay inside)
        p += __shfl_xor(p, 1, 32);
        p += __shfl_xor(p, 2, 32);
        p += __shfl_xor(p, 4, 32);
        p += __shfl_xor(p, 8, 32);
        rinv[r] = 1.0f / p;
    }

    float* Orow = out + (size_t)bh * NSEQ * DIM + (size_t)(n0 + 8 * hi) * DIM;
    #pragma unroll
    for (int r = 0; r < 8; ++r) {
        #pragma unroll
        for (int ct = 0; ct < 4; ++ct) {
            Orow[(size_t)r * DIM + ct * 16 + l15] = e[ct][r] * cx[ct] * rinv[r];
        }
    }
}

// ---------------------------------------------------------------------------
extern "C" void kernel_launch(void* const* d_in, const int* in_sizes, int n_in,
                              void* d_out, int out_size, void* d_ws, size_t ws_size,
                              hipStream_t stream) {
    const float* q    = (const float*)d_in[0];
    const float* k    = (const float*)d_in[1];
    const float* v    = (const float*)d_in[2];
    const float* proj = (const float*)d_in[3];
    float* out = (float*)d_out;

    float* ws_ksum = (float*)d_ws;                 // [32][64]
    float* ws_ctx  = ws_ksum + BH * DIM;           // [32][64]

    const int ws_elems = 2 * BH * DIM;             // 4096 floats
    FA_zero_ws<<<(ws_elems + 255) / 256, 256, 0, stream>>>(ws_ksum, ws_elems);

    FA_pass1<<<dim3(BH, NSEQ / CHUNK), 256, 0, stream>>>(
        k, v, proj, ws_ksum, ws_ctx);

    FA_pass2<<<dim3(BH, NSEQ / 128), 256, 0, stream>>>(
        q, proj, ws_ksum, ws_ctx, out);
}